// CovFunction_16166256902674
// MI455X (gfx1250) — compile-verified
//
#include <hip/hip_runtime.h>
#include <hip/hip_bf16.h>

typedef __attribute__((ext_vector_type(2))) float v2f;
typedef __attribute__((ext_vector_type(8))) float v8f;

#define NROWS 8192
#define MCOLS 8192
#define DK    16
#define JT    4          // j-tiles per wave

#define LOG2E 1.4426950408889634f

// One wave32 computes four adjacent 16x16 output tiles (same i-tile) of
// K = exp(2*sn - 0.5*sq) using V_WMMA_F32_16X16X4_F32 (exact f32 GEMM path).
//
// f32 A (16x4) fragment layout (ISA 7.12.2): lane l holds row m = l&15;
// VGPR0 = K = 2*(l>=16), VGPR1 = K+1. B (4x16) = Vs^T has the identical
// float2 pattern on the Vs row. C/D: VGPR r -> M = r + 8*(l>=16), N = l&15.
//
// Per tile: 4 dot WMMAs accumulate L*sum_k s_k^2 u_k v_k (A pre-scaled by
// L = log2e), then a 5th WMMA folds in the rank-1 norm corrections AND the
// cross-half-wave reduction of the per-lane norm partials:
//   slot0: A=L*c2 - (0.5/L)*upart_h0, B=1      slot1: A=L, B=-0.5*vpart_h0
//   slot2: A=     - (0.5/L)*upart_h1, B=1      slot3: A=L, B=-0.5*vpart_h1
// giving acc = L*(c2 - 0.5*||Us_i||^2 - 0.5*||Vs_j||^2 + Us_i.Vs_j), so the
// finish is exp2(min(acc, L*c2)) per element (min == reference's max(sq,0)).
// A-side work (w-exps, A scaling, upart) is amortized over all 4 j-tiles.
__global__ __launch_bounds__(256) void ard_rbf_wmma_kernel(
    const float* __restrict__ U,    // (8192, 16)
    const float* __restrict__ V,    // (8192, 16)
    const float* __restrict__ w,    // (16, 1)
    const float* __restrict__ sn,   // scalar
    float* __restrict__ out)        // (8192, 8192)
{
    const int lane = threadIdx.x & 31;
    const int wave = threadIdx.x >> 5;
    const int tj0  = (blockIdx.x * 8 + wave) * JT; // first of JT column tiles
    const int ti   = blockIdx.y;                   // row tile [0, 512)

    const int half  = lane >> 4;                   // 0 or 1
    const int l15   = lane & 15;
    const int kbase = half * 2;                    // K sub-offset: 0 or 2

    const float* urow = U + (size_t)(ti * 16 + l15) * DK;
    const float* vrow[JT];
#pragma unroll
    for (int t = 0; t < JT; ++t)
        vrow[t] = V + (size_t)((tj0 + t) * 16 + l15) * DK;

    v2f a[4], b[JT][4];
    float upart = 0.0f;
    float vp[JT] = {};
#pragma unroll
    for (int kk = 0; kk < 4; ++kk) {
        const int k0 = 4 * kk + kbase;
        const float s0 = __builtin_amdgcn_exp2f(-LOG2E * w[k0]);     // exp(-w)
        const float s1 = __builtin_amdgcn_exp2f(-LOG2E * w[k0 + 1]);
        const float a0 = urow[k0]     * (s0 * LOG2E);  // A carries the log2e
        const float a1 = urow[k0 + 1] * (s1 * LOG2E);
        v2f av = {a0, a1};
        a[kk] = av;
        upart += a0 * a0 + a1 * a1;    // = L^2 * (this half's ||Us_row||^2)
#pragma unroll
        for (int t = 0; t < JT; ++t) {
            const float b0 = vrow[t][k0] * s0;
            const float b1 = vrow[t][k0 + 1] * s1;
            v2f bv = {b0, b1};
            b[t][kk] = bv;
            vp[t] += b0 * b0 + b1 * b1;  // this half's ||Vs_col||^2
        }
    }

    v8f acc[JT];
#pragma unroll
    for (int t = 0; t < JT; ++t) {
        v8f z = {};
        acc[t] = z;
    }
#pragma unroll
    for (int kk = 0; kk < 4; ++kk) {
#pragma unroll
        for (int t = 0; t < JT; ++t) {
            acc[t] = __builtin_amdgcn_wmma_f32_16x16x4_f32(
                false, a[kk], false, b[t][kk], (short)0, acc[t], false, false);
        }
    }

    // 5th WMMA per tile: rank-1 norm/bias corrections + cross-half reduction.
    const float c2l = (2.0f * LOG2E) * sn[0];          // log2e * 2*sn
    const float hf  = (half == 0) ? 1.0f : 0.0f;
    v2f ae = { hf * c2l - (0.5f / LOG2E) * upart, LOG2E };
#pragma unroll
    for (int t = 0; t < JT; ++t) {
        v2f be = { 1.0f, -0.5f * vp[t] };
        acc[t] = __builtin_amdgcn_wmma_f32_16x16x4_f32(
            false, ae, false, be, (short)0, acc[t], false, false);
    }

    float* ob = out + (size_t)(ti * 16) * MCOLS + (size_t)(tj0 * 16 + l15);
#pragma unroll
    for (int r = 0; r < 8; ++r) {
        const int m = r + half * 8;
        float* orow = ob + (size_t)m * MCOLS;
#pragma unroll
        for (int t = 0; t < JT; ++t) {
            const float v = __builtin_amdgcn_exp2f(fminf(acc[t][r], c2l));
            // 256 MB streaming output: non-temporal, don't evict Us/Vs in L2.
            __builtin_nontemporal_store(v, orow + 16 * t);
        }
    }
}

extern "C" void kernel_launch(void* const* d_in, const int* in_sizes, int n_in,
                              void* d_out, int out_size, void* d_ws, size_t ws_size,
                              hipStream_t stream) {
    const float* U  = (const float*)d_in[0];
    const float* V  = (const float*)d_in[1];
    const float* w  = (const float*)d_in[2];
    const float* sn = (const float*)d_in[3];
    float* out = (float*)d_out;

    // 512x512 tiles of 16x16; 8 waves x 4 j-tiles per 256-thread block.
    dim3 grid(MCOLS / (16 * 8 * JT), NROWS / 16);  // (16, 512)
    ard_rbf_wmma_kernel<<<grid, dim3(256), 0, stream>>>(U, V, w, sn, out);
}